// C4Transformer_62380105007538
// MI455X (gfx1250) — compile-verified
//
#include <hip/hip_runtime.h>
#include <stdint.h>

// C4Transformer expert-mixture kernel for MI455X (gfx1250, wave32).
//
// Memory floor ~0.7us (16MB @ 23.3TB/s). Per-element math reduced to
// ~5 transcendentals via: (a) closed-form windowed softmax (one exp2,
// constant tap coefficients, recip[kc+j] factored as recip[kc]*2^(-j/8)),
// (b) silu(-t)=silu(t)-t sharing for all six comparison gates (3 exp2),
// (c) opcode-router weights as a conflict-free 64-entry LDS LUT gather.
// Recip table staged to LDS with gfx1250 async global->LDS DMA.

namespace {

constexpr int   NK  = 161;
constexpr float L2E = 1.44269504088896340736f;  // log2(e)

__device__ __forceinline__ float fexp2(float x) { return __builtin_amdgcn_exp2f(x); }
__device__ __forceinline__ float frcp (float x) { return __builtin_amdgcn_rcpf(x);  }
__device__ __forceinline__ float flog2(float x) { return __builtin_amdgcn_logf(x);  }

__device__ __forceinline__ float silu(float t) {
  return t * frcp(1.0f + fexp2(-L2E * t));
}
__device__ __forceinline__ float sharp_gate(float x) {
  return (silu(20.0f * x + 10.0f) - silu(20.0f * x - 10.0f)) * 0.05f;
}
__device__ __forceinline__ float eq_gate_f(float d) {
  return sharp_gate(d + 0.5f) * sharp_gate(0.5f - d);
}

typedef float v4f __attribute__((ext_vector_type(4)));
typedef int   v4i __attribute__((ext_vector_type(4)));

__device__ __forceinline__ float compute_one(float av, float bv, int op,
                                             const float* s_recip,
                                             const float* s_w) {
  // ---- LogDivision: closed-form windowed softmax over 2^-k/8 table ----
  // weights relative to nearest key: w(+m)=Ep*r^m, w(-m)=En*r^m, w(0)=1,
  // r = exp(-6.25); numerator taps fold the 2^(-+m/8) recip factors.
  const bool  bpos = bv > 0.0f;
  const float Lb   = flog2(fabsf(bpos ? bv : 1.0f));
  float kcf = rintf(Lb * 8.0f);
  kcf = fminf(fmaxf(kcf, 0.0f), 160.0f);
  const int   kc = (int)kcf;
  const float f  = fmaf(kcf, -0.125f, Lb);            // Lb - kc/8
  const float E  = fexp2(144.26950409f * fabsf(f));   // exp(100*|f|)
  const float Ep = (f >= 0.0f) ? E : 1.0f;            // applies to +m taps
  const float En = (f >= 0.0f) ? 1.0f : E;            // applies to -m taps
  constexpr float DN[4] = {1.9304541e-3f, 3.7266532e-6f, 7.1941332e-9f, 1.3887946e-11f};
  constexpr float NP[4] = {1.7702342e-3f, 3.1337294e-6f, 5.5474353e-9f, 9.8202570e-12f};
  constexpr float NN[4] = {2.1051751e-3f, 4.4317625e-6f, 9.3296365e-9f, 1.9640522e-11f};
  float num = 1.0f, den = 1.0f;
#pragma unroll
  for (int m = 1; m <= 4; ++m) {
    if (kc <= NK - 1 - m) { num += Ep * NP[m - 1]; den += Ep * DN[m - 1]; }
    if (kc >= m)          { num += En * NN[m - 1]; den += En * DN[m - 1]; }
  }
  const float recip = s_recip[kc] * num * frcp(den);
  const float div_  = bpos ? floorf(av * recip) : 0.0f;
  const float mod_  = (bv == 0.0f) ? 0.0f : fmaf(-div_, bv, av);

  // ---- BitwiseExpert ----
  const int ai = (int)av;
  const int bi = (int)bv;
  const int sh = min(max(bi, 0), 31);
  const float or_  = (float)(ai | bi);
  const float xor_ = (float)(ai ^ bi);
  const float and_ = (float)(ai & bi);
  const float shl_ = (float)(ai << sh);
  const float shr_ = (float)(ai >> sh);

  // ---- ComparisonExpert: 3 shared silu evals ----
  const float t2 = 20.0f * (av - bv);
  const float A  = silu(t2 + 20.0f);
  const float B  = silu(t2);
  const float C  = silu(t2 - 20.0f);
  const float ge = (A - B) * 0.05f;
  const float gt = (B - C) * 0.05f;
  const float le = 1.0f - gt;
  const float lt = 1.0f - ge;
  const float eq = ge * le;
  const float ne = 1.0f - eq;

  // ---- Gated sum over experts 14..29 via LDS weight LUT ----
  const float R[16] = { or_, xor_, and_, eq, ne, lt, gt, le, ge,
                        shl_, shr_, av + bv, av - bv, av * bv, div_, mod_ };
  // weight for expert e=14+j is s_w[32 + op - 14 - j] = wb[15 - j]
  const float* wb = s_w + (op + 3);
  float acc = 0.0f;
#pragma unroll
  for (int j = 0; j < 16; ++j) acc = fmaf(wb[15 - j], R[j], acc);
  return acc;
}

__global__ __launch_bounds__(256) void c4_moe_kernel(
    const float* __restrict__ a, const float* __restrict__ b,
    const float* __restrict__ recip_tab, const int* __restrict__ opcode,
    float* __restrict__ out, int n) {
  __shared__ float s_recip[192];  // 161 used
  __shared__ float s_w[64];       // router weight LUT, one entry per bank

  const int t = threadIdx.x;
  // Stage reciprocal table via CDNA5 async global->LDS DMA.
#if defined(__HIP_DEVICE_COMPILE__) && defined(__gfx1250__)
  if (t < NK) {
    const uint32_t lds_addr = (uint32_t)(uintptr_t)(&s_recip[t]);
    const uint64_t gaddr    = (uint64_t)(uintptr_t)(recip_tab + t);
    asm volatile("global_load_async_to_lds_b32 %0, %1, off"
                 :: "v"(lds_addr), "v"(gaddr) : "memory");
  }
#else
  if (t < NK) s_recip[t] = recip_tab[t];
#endif
  // Router gates depend only on |op - e|: 4 values, rest ~0 (<1e-17).
  if (t < 64) {
    const int dt = t - 32;
    const int ad = dt < 0 ? -dt : dt;
    const float g = (ad == 0) ? eq_gate_f(0.0f)
                  : (ad == 1) ? eq_gate_f(1.0f)
                  : (ad == 2) ? eq_gate_f(2.0f)
                  : (ad == 3) ? eq_gate_f(3.0f) : 0.0f;
    s_w[t] = g;
  }
#if defined(__HIP_DEVICE_COMPILE__) && defined(__gfx1250__)
  asm volatile("s_wait_asynccnt 0" ::: "memory");
#endif
  __syncthreads();

  const int base = (blockIdx.x * blockDim.x + threadIdx.x) * 4;
  if (base + 3 < n) {
    const v4f a4 = __builtin_nontemporal_load((const v4f*)(a + base));
    const v4f b4 = __builtin_nontemporal_load((const v4f*)(b + base));
    const v4i o4 = __builtin_nontemporal_load((const v4i*)(opcode + base));
    v4f r;
    r.x = compute_one(a4.x, b4.x, o4.x, s_recip, s_w);
    r.y = compute_one(a4.y, b4.y, o4.y, s_recip, s_w);
    r.z = compute_one(a4.z, b4.z, o4.z, s_recip, s_w);
    r.w = compute_one(a4.w, b4.w, o4.w, s_recip, s_w);
    __builtin_nontemporal_store(r, (v4f*)(out + base));
  } else {
    for (int i = base; i < n; ++i)
      out[i] = compute_one(a[i], b[i], opcode[i], s_recip, s_w);
  }
}

}  // namespace

extern "C" void kernel_launch(void* const* d_in, const int* in_sizes, int n_in,
                              void* d_out, int out_size, void* d_ws, size_t ws_size,
                              hipStream_t stream) {
  // setup_inputs order: a, b, log_keys, recip_values, opcode
  const float* a     = (const float*)d_in[0];
  const float* b     = (const float*)d_in[1];
  const float* recip = (const float*)d_in[3];
  const int*   op    = (const int*)d_in[4];
  float*       out   = (float*)d_out;
  const int n = in_sizes[0];

  constexpr int TPB = 256, EPT = 4;
  const int grid = (n + TPB * EPT - 1) / (TPB * EPT);
  hipLaunchKernelGGL(c4_moe_kernel, dim3(grid), dim3(TPB), 0, stream,
                     a, b, recip, op, out, n);
}